// SparseAutoencoder_20005957664922
// MI455X (gfx1250) — compile-verified
//
#include <hip/hip_runtime.h>
#include <hip/hip_bf16.h>

typedef __attribute__((ext_vector_type(2))) float    v2f;
typedef __attribute__((ext_vector_type(8))) float    v8f;
typedef __attribute__((ext_vector_type(4))) unsigned v4u;
typedef __attribute__((ext_vector_type(8))) int      v8i;
typedef __attribute__((ext_vector_type(4))) int      v4i;

#define M_ROWS 4096
#define K_DIM  768
#define N_DIM  16384
#define TOPK   32

// ---------------------------------------------------------------------------
// Kernel 0: inverse row norms of W  ->  inv_norm[16384] in workspace
// ---------------------------------------------------------------------------
__global__ __launch_bounds__(256) void norms_kernel(const float* __restrict__ W,
                                                    float* __restrict__ inv_norm) {
    const int lane = threadIdx.x & 31;
    const int wave = threadIdx.x >> 5;
    const int row  = blockIdx.x * 8 + wave;
    if (row >= N_DIM) return;
    const float* wr = W + (size_t)row * K_DIM;
    float s = 0.f;
#pragma unroll
    for (int i = 0; i < K_DIM / 32; ++i) {
        float v = wr[lane + 32 * i];
        s += v * v;
    }
#pragma unroll
    for (int off = 16; off > 0; off >>= 1)
        s += __shfl_xor(s, off, 32);
    if (lane == 0) inv_norm[row] = rsqrtf(s);
}

// ---------------------------------------------------------------------------
// Tensor Data Mover helpers: DMA a rows x 16-float tile (row stride 768 f32)
// from global into LDS, padding 2 DWORDs after every 16 DWORDs so the LDS
// row stride is 18 floats (bank-conflict-free, matches compute indexing).
// D# bitfields per CDNA5 ISA.
// ---------------------------------------------------------------------------
__device__ __forceinline__ unsigned lds_off(const void* p) {
    // generic LDS address: low 32 bits are the LDS byte offset
    return (unsigned)(unsigned long long)p;
}

__device__ __forceinline__ void tdm_load_tile(const float* gptr, unsigned ldsaddr,
                                              unsigned tileRows, unsigned tensorRows,
                                              unsigned tensorCols) {
    const unsigned long long ga = (unsigned long long)gptr;

    v4u g0;
    g0.x = 1u;                                    // count=1 (valid), user mode
    g0.y = ldsaddr;                               // lds_addr (bytes)
    g0.z = (unsigned)(ga & 0xffffffffu);          // global_addr[31:0]
    g0.w = (unsigned)(ga >> 32) | 0x80000000u;    // global_addr[56:32] | type=2

    const unsigned d0 = (2u << 16)                // data_size = 4 bytes
                      | (1u << 20)                // pad_enable
                      | (3u << 22)                // pad_interval: 16 DWORDs
                      | (1u << 25);               // pad_amount: 2 DWORDs
    v8i g1;
    g1[0] = (int)d0;                                            // wg_mask=0 | flags
    g1[1] = (int)((tensorCols & 0xffffu) << 16);                // dim0[15:0]
    g1[2] = (int)((tensorCols >> 16) | ((tensorRows & 0xffffu) << 16)); // dim0.hi|dim1.lo
    g1[3] = (int)((tensorRows >> 16) | (16u << 16));            // dim1.hi | tile_dim0=16
    g1[4] = (int)(tileRows & 0xffffu);                          // tile_dim1 | tile_dim2=0
    g1[5] = (int)K_DIM;                                         // tensor_dim0_stride lo32
    g1[6] = 0;                                                  // stride0.hi | stride1.lo
    g1[7] = 0;                                                  // stride1.hi

    const v4i z4 = {0, 0, 0, 0};
    const v8i z8 = {0, 0, 0, 0, 0, 0, 0, 0};
    __builtin_amdgcn_tensor_load_to_lds(g0, g1, z4, z4, z8, 0);
}

// ---------------------------------------------------------------------------
// Kernel 1: pre = x @ W^T + b via V_WMMA_F32_16X16X4_F32.
// Block tile 64(M) x 128(N), 8 waves in a 4(M) x 2(N) grid; each wave owns a
// 16x64 strip = four 16x16 f32 accumulators (A fragment reused 4x).
// K staged in 16-wide slabs, double-buffered in LDS, filled by the TDM
// (wave 0 issues tensor_load_to_lds; s_wait_tensorcnt + barrier publishes).
// ---------------------------------------------------------------------------
__global__ __launch_bounds__(256) void gemm_pre_kernel(const float* __restrict__ x,
                                                       const float* __restrict__ W,
                                                       const float* __restrict__ bias,
                                                       float* __restrict__ enc) {
    __shared__ float lx[2][64][18];    // x tile:  64 rows x 16 K (stride 18)
    __shared__ float lw[2][128][18];   // W tile: 128 rows x 16 K (stride 18)

    const int tid   = threadIdx.x;
    const int lane  = tid & 31;
    const int wave  = tid >> 5;
    const int waveM = wave >> 1;            // 0..3 -> M offset 16*waveM
    const int waveN = wave & 1;             // 0..1 -> N offset 64*waveN
    const int gmBase = blockIdx.y * 64;
    const int gnBase = blockIdx.x * 128;

    v8f c[4] = {{}, {}, {}, {}};

    const int nIter = K_DIM / 16;           // 48

    if (wave == 0) {
        tdm_load_tile(x + (size_t)gmBase * K_DIM, lds_off(&lx[0][0][0]),
                      64u, (unsigned)(M_ROWS - gmBase), (unsigned)K_DIM);
        tdm_load_tile(W + (size_t)gnBase * K_DIM, lds_off(&lw[0][0][0]),
                      128u, (unsigned)(N_DIM - gnBase), (unsigned)K_DIM);
    }

    for (int it = 0; it < nIter; ++it) {
        const int buf = it & 1;
        if (wave == 0) {
            if (it + 1 < nIter) {
                const int k0n = (it + 1) * 16;
                tdm_load_tile(x + (size_t)gmBase * K_DIM + k0n, lds_off(&lx[buf ^ 1][0][0]),
                              64u, (unsigned)(M_ROWS - gmBase), (unsigned)(K_DIM - k0n));
                tdm_load_tile(W + (size_t)gnBase * K_DIM + k0n, lds_off(&lw[buf ^ 1][0][0]),
                              128u, (unsigned)(N_DIM - gnBase), (unsigned)(K_DIM - k0n));
                __builtin_amdgcn_s_wait_tensorcnt(2);   // current buf's 2 loads done
            } else {
                __builtin_amdgcn_s_wait_tensorcnt(0);
            }
        }
        __syncthreads();    // publish tile[buf] to all waves

#pragma unroll
        for (int kk = 0; kk < 16; kk += 4) {
            const int kb = kk + (lane >> 4) * 2;       // K=0/1 half vs K=2/3 half
            const int ar = waveM * 16 + (lane & 15);
            v2f a;
            a.x = lx[buf][ar][kb];
            a.y = lx[buf][ar][kb + 1];
#pragma unroll
            for (int t = 0; t < 4; ++t) {
                const int bc = waveN * 64 + t * 16 + (lane & 15);
                v2f b;
                b.x = lw[buf][bc][kb];
                b.y = lw[buf][bc][kb + 1];
                c[t] = __builtin_amdgcn_wmma_f32_16x16x4_f32(false, a, false, b,
                                                             (short)0, c[t], false, false);
            }
        }
        __syncthreads();    // all waves done reading buf before TDM overwrites it
    }

    // Epilogue: add bias, store pre into the encoded region.
    const int mHalf = (lane >> 4) * 8;
#pragma unroll
    for (int t = 0; t < 4; ++t) {
        const int gn  = gnBase + waveN * 64 + t * 16 + (lane & 15);
        const float bb = bias[gn];
#pragma unroll
        for (int i = 0; i < 8; ++i) {
            const size_t gm = (size_t)(gmBase + waveM * 16 + mHalf + i);
            enc[gm * N_DIM + gn] = c[t][i] + bb;
        }
    }
}

// ---------------------------------------------------------------------------
// Kernel 2: per-row exact top-K threshold (4-pass MSB radix select),
// in-place thresholding, deterministic sparse decode.
// ---------------------------------------------------------------------------
__device__ __forceinline__ unsigned f2key(float f) {
    unsigned u = __float_as_uint(f);
    return (u & 0x80000000u) ? ~u : (u | 0x80000000u);
}
__device__ __forceinline__ float key2f(unsigned k) {
    unsigned u = (k & 0x80000000u) ? (k & 0x7fffffffu) : ~k;
    return __uint_as_float(u);
}

#define NNZ_CAP 256

__global__ __launch_bounds__(256) void select_decode_kernel(const float* __restrict__ W,
                                                            const float* __restrict__ inv_norm,
                                                            float* __restrict__ enc,
                                                            float* __restrict__ dec) {
    __shared__ unsigned hist[256];
    __shared__ unsigned s_prefix;
    __shared__ int      s_remaining;
    __shared__ int      s_nnz;
    __shared__ int      sidx[NNZ_CAP];
    __shared__ float    sval[NNZ_CAP];

    const int row = blockIdx.x;
    const int tid = threadIdx.x;
    float* pre = enc + (size_t)row * N_DIM;

    if (tid == 0) { s_prefix = 0u; s_remaining = TOPK; s_nnz = 0; }

    for (int p = 3; p >= 0; --p) {
        hist[tid] = 0u;
        __syncthreads();
        const unsigned prefix = s_prefix;
        for (int j = tid; j < N_DIM; j += 256) {
            const unsigned key = f2key(pre[j]);
            const bool match = (p == 3) || ((key >> ((p + 1) * 8)) == prefix);
            if (match) atomicAdd(&hist[(key >> (p * 8)) & 255u], 1u);
        }
        __syncthreads();
        if (tid == 0) {
            int rem = s_remaining;
            unsigned c = 0;
            for (int bin = 255; bin >= 0; --bin) {
                c += hist[bin];
                if ((int)c >= rem) {
                    s_remaining = rem - (int)(c - hist[bin]);
                    s_prefix = (s_prefix << 8) | (unsigned)bin;
                    break;
                }
            }
        }
        __syncthreads();
    }
    const float thresh = key2f(s_prefix);

    for (int j = tid; j < N_DIM; j += 256) {
        const float f = pre[j];
        const bool keep = (f >= thresh);
        pre[j] = keep ? f : 0.0f;
        if (keep) {
            const int pos = atomicAdd(&s_nnz, 1);
            if (pos < NNZ_CAP) {
                sidx[pos] = j;
                sval[pos] = f * inv_norm[j];
            }
        }
    }
    __syncthreads();

    const int nn = (s_nnz < NNZ_CAP) ? s_nnz : NNZ_CAP;

    if (tid == 0) {   // sort by column index -> deterministic f32 sum order
        for (int i = 1; i < nn; ++i) {
            const int   ki = sidx[i];
            const float kv = sval[i];
            int j = i - 1;
            while (j >= 0 && sidx[j] > ki) {
                sidx[j + 1] = sidx[j];
                sval[j + 1] = sval[j];
                --j;
            }
            sidx[j + 1] = ki;
            sval[j + 1] = kv;
        }
    }
    __syncthreads();

    float a0 = 0.f, a1 = 0.f, a2 = 0.f;
    for (int s = 0; s < nn; ++s) {
        const float v = sval[s];
        const float* wr = W + (size_t)sidx[s] * K_DIM;
        a0 += v * wr[tid];
        a1 += v * wr[tid + 256];
        a2 += v * wr[tid + 512];
    }
    float* dr = dec + (size_t)row * K_DIM;
    dr[tid]       = a0;
    dr[tid + 256] = a1;
    dr[tid + 512] = a2;
}

// ---------------------------------------------------------------------------
extern "C" void kernel_launch(void* const* d_in, const int* in_sizes, int n_in,
                              void* d_out, int out_size, void* d_ws, size_t ws_size,
                              hipStream_t stream) {
    const float* x    = (const float*)d_in[0];   // 4096 x 768
    const float* W    = (const float*)d_in[1];   // 16384 x 768
    const float* bias = (const float*)d_in[2];   // 16384

    float* dec = (float*)d_out;                          // 4096 x 768
    float* enc = dec + (size_t)M_ROWS * K_DIM;           // 4096 x 16384
    float* inv_norm = (float*)d_ws;                      // 16384 floats

    (void)in_sizes; (void)n_in; (void)out_size; (void)ws_size;

    norms_kernel<<<N_DIM / 8, 256, 0, stream>>>(W, inv_norm);

    dim3 ggrid(N_DIM / 128, M_ROWS / 64);
    gemm_pre_kernel<<<ggrid, 256, 0, stream>>>(x, W, bias, enc);

    select_decode_kernel<<<M_ROWS, 256, 0, stream>>>(W, inv_norm, enc, dec);
}